// BetterPixelBilateralFilter2_75926431858855
// MI455X (gfx1250) — compile-verified
//
#include <hip/hip_runtime.h>
#include <stdint.h>

// ---------------------------------------------------------------------------
// Bilateral filter, MI455X (gfx1250).
//   * 64x16 output tile per 1024-thread block, 1 pixel/thread (32 waves/WGP).
//   * Channel loop OUTERMOST and rolled: no per-channel register arrays
//     anywhere (rounds 3-5 showed arrays being demoted to scratch when tap
//     loops didn't unroll). 12 packed v2f tap-pair accumulators instead.
//   * Packed fp32 (v_pk_fma_f32, probe-verified) via tap-symmetry pairing:
//     row pairs (-dx,+dx) and column pairs (-2,-1)/(+1,+2) share scalar
//     pc/m2b broadcasts; the (0,0) tap is exact w=1 (reference d2 == 0).
//   * LDS halo tile 76x28 x 32ch = 272,384 B ([y][c][x]); interior tiles
//     DMA'd by ONE tensor_load_to_lds (TDM), completed by s_wait_tensorcnt.
//     Border tiles: manual fill. Scale reads prefetched (global_prefetch)
//     before the barrier to overlap the DMA.
// ---------------------------------------------------------------------------

namespace {
constexpr int kNB   = 2;
constexpr int kCin  = 66;   // 2*C + 2
constexpr int kC    = 32;
constexpr int kH    = 720;
constexpr int kW    = 1280;
constexpr int kTW   = 64;   // tile width in pixels
constexpr int kTH   = 16;
constexpr int kBX   = 64;
constexpr int kBY   = 16;
constexpr int kHalo = 6;                  // (kernel_size/2) * dilation = 2*3
constexpr int kLW   = kTW + 2 * kHalo;    // 76
constexpr int kLH   = kTH + 2 * kHalo;    // 28
constexpr int kRow  = kC * kLW;           // 2432 floats per y-row in LDS
constexpr int kTileElems = kLH * kRow;    // 68,096 floats = 272,384 B
}

typedef float v2f __attribute__((ext_vector_type(2)));
typedef __attribute__((ext_vector_type(4))) unsigned int u32x4_t;
typedef __attribute__((ext_vector_type(8))) int          i32x8_t;
typedef __attribute__((ext_vector_type(4))) int          i32x4_t;

typedef __attribute__((address_space(3))) float lds_f_t;

#if defined(__HIP_DEVICE_COMPILE__) && __has_builtin(__builtin_amdgcn_tensor_load_to_lds)
#define HAVE_TDM 1
#else
#define HAVE_TDM 0
#endif

__device__ __forceinline__ float softplus_f(float s) {
  return (s > 15.f) ? s : __logf(1.f + __expf(s));
}

__launch_bounds__(kBX * kBY, 1)
__global__ void bilateral_kernel(const float* __restrict__ in,
                                 const float* __restrict__ ecoef,
                                 float* __restrict__ out) {
  __shared__ __align__(16) float s_tile[kTileElems];  // [y][c][x]

  const int tx = threadIdx.x, ty = threadIdx.y;
  const int bx = blockIdx.x,  by = blockIdx.y, bb = blockIdx.z;
  const int gx0 = bx * kTW,   gy0 = by * kTH;
  const int gx  = gx0 + tx;
  const int gy  = gy0 + ty;
  const bool interior = (bx >= 1) && (bx <= (int)gridDim.x - 2) &&
                        (by >= 1) && (by <= (int)gridDim.y - 2);

  const size_t plane = (size_t)kH * kW;
  const float* inb = in + (size_t)bb * kCin * plane;

  const uint32_t tile_base = (uint32_t)(uintptr_t)&s_tile[0];
  const lds_f_t* t3 = (const lds_f_t*)tile_base;

  // ------------------------------------------------------------------ fill
  bool use_tdm = false;
#if HAVE_TDM
  use_tdm = interior;
  if (use_tdm) {
    if (tx == 0 && ty == 0) {
      uint64_t ga = (uint64_t)(uintptr_t)(inb + (size_t)(gy0 - kHalo) * kW +
                                          (gx0 - kHalo));
      uint32_t ga_lo = (uint32_t)__builtin_amdgcn_readfirstlane((int)(uint32_t)ga);
      uint32_t ga_hi = (uint32_t)__builtin_amdgcn_readfirstlane((int)(uint32_t)(ga >> 32));
      uint32_t lds_off = (uint32_t)__builtin_amdgcn_readfirstlane((int)tile_base);

      // ---- D# group 0: count=1 | lds_addr | global_addr | type=2 ----
      u32x4_t g0 = { 1u, lds_off, ga_lo,
                     (ga_hi & 0x01FFFFFFu) | (2u << 30) };
      // ---- D# group 1: dim0=x(76/kW), dim1=c(32, stride plane),
      //                  dim2=y(28, stride kW) -> LDS gets [y][c][x] ----
      i32x8_t g1 = { (int)0x00020000,
                     (int)(((uint32_t)kW & 0xFFFFu) << 16),
                     (int)(((uint32_t)kC & 0xFFFFu) << 16),
                     (int)((uint32_t)kLW << 16),
                     (int)((uint32_t)kC | ((uint32_t)kLH << 16)),
                     (int)0x000E1000,                        // stride0 = 921600
                     (int)(((uint32_t)kW & 0xFFFFu) << 16),  // stride1 = 1280
                     (int)0 };
      i32x4_t g2 = { (int)kH, 1, 0, 0 };
      i32x4_t g3 = { 0, 0, 0, 0 };
#if defined(__clang_major__) && __clang_major__ >= 23
      i32x8_t gz = { 0, 0, 0, 0, 0, 0, 0, 0 };
      __builtin_amdgcn_tensor_load_to_lds(g0, g1, g2, g3, gz, 0);
#else
      __builtin_amdgcn_tensor_load_to_lds(g0, g1, g2, g3, 0);
#endif
    }
  }
#endif  // HAVE_TDM

  if (!use_tdm) {
    lds_f_t* w3 = (lds_f_t*)tile_base;
    const int tid = ty * kBX + tx;
    for (int idx = tid; idx < kTileElems; idx += kBX * kBY) {
      int y = idx / kRow;
      int r = idx - y * kRow;
      int c = r / kLW;
      int x = r - c * kLW;
      int sy = gy0 - kHalo + y;
      int sx = gx0 - kHalo + x;
      float v = 0.f;
      if ((unsigned)sy < (unsigned)kH && (unsigned)sx < (unsigned)kW)
        v = inb[(size_t)c * plane + (size_t)sy * kW + sx];
      w3[idx] = v;
    }
  }

  // Prefetch this pixel's 34 scale channels (overlaps the TDM DMA / barrier).
  const float* sc = inb + (size_t)kC * plane + (size_t)gy * kW + gx;
#pragma unroll
  for (int c = 0; c < kC + 2; ++c)
    __builtin_prefetch(sc + (size_t)c * plane, 0, 3);

  const float psy = (-ecoef[kC])     * softplus_f(sc[(size_t)kC * plane]);
  const float psx = (-ecoef[kC + 1]) * softplus_f(sc[(size_t)(kC + 1) * plane]);

#if HAVE_TDM
#if __has_builtin(__builtin_amdgcn_s_wait_tensorcnt)
  __builtin_amdgcn_s_wait_tensorcnt(0);
#else
  asm volatile("s_wait_tensorcnt 0" ::: "memory");
#endif
#endif
  __syncthreads();

  // ---------------------------------------------------------------- bases
  // Biased row bases: [0]=lx-6 [3]=lx-3 [6]=lx [9]=lx+3 [12]=lx+6 (dwords).
  const int lxc = tx + kHalo;
  const int lyc = ty + kHalo;
  const lds_f_t* bC = t3 + lyc * kRow + (lxc - 6);
  const lds_f_t* bA = bC - 6 * kRow;   // dy = -2
  const lds_f_t* bB = bC - 3 * kRow;   // dy = -1
  const lds_f_t* bD = bC + 3 * kRow;   // dy = +1
  const lds_f_t* bE = bC + 6 * kRow;   // dy = +2

  // ------------------------------------- channel-major accumulation loop
  // 12 packed tap pairs: a0..a9 = row pairs (k=2,k=1 per row), a10/a11 = cols.
  v2f a0={0,0}, a1={0,0}, a2={0,0}, a3={0,0}, a4={0,0}, a5={0,0};
  v2f a6={0,0}, a7={0,0}, a8={0,0}, a9={0,0}, a10={0,0}, a11={0,0};
  float S = 0.f;

#define ACC_PAIR(ACC, L, R)                                                   \
  { v2f nb = { (L), (R) };                                                    \
    v2f t = __builtin_elementwise_fma(pcs, nb, mbs);                          \
    (ACC) = __builtin_elementwise_fma(t, nb, (ACC)); }

  {
    const float* scp = sc;
#pragma unroll 4
    for (int c = 0; c < kC; ++c) {
      const float pc = (-ecoef[c]) * softplus_f(*scp);
      scp += plane;
      const int coff = c * kLW;
      const float xv = bC[coff + 6];
      const float px = pc * xv;
      const float mb = -2.f * px;
      S = __builtin_fmaf(px, xv, S);
      const v2f pcs = { pc, pc };
      const v2f mbs = { mb, mb };
      ACC_PAIR(a0, bA[coff],     bA[coff + 12])   // (-2,-2)/(-2,+2)
      ACC_PAIR(a1, bA[coff + 3], bA[coff + 9])    // (-2,-1)/(-2,+1)
      ACC_PAIR(a2, bB[coff],     bB[coff + 12])
      ACC_PAIR(a3, bB[coff + 3], bB[coff + 9])
      ACC_PAIR(a4, bC[coff],     bC[coff + 12])
      ACC_PAIR(a5, bC[coff + 3], bC[coff + 9])
      ACC_PAIR(a6, bD[coff],     bD[coff + 12])
      ACC_PAIR(a7, bD[coff + 3], bD[coff + 9])
      ACC_PAIR(a8, bE[coff],     bE[coff + 12])
      ACC_PAIR(a9, bE[coff + 3], bE[coff + 9])
      ACC_PAIR(a10, bA[coff + 6], bB[coff + 6])   // (-2,0)/(-1,0)
      ACC_PAIR(a11, bD[coff + 6], bE[coff + 6])   // (+1,0)/(+2,0)
    }
  }
#undef ACC_PAIR

  // ------------------------------------------------ finale: 12 exps + sums
  v2f num0 = {0.f,0.f}, num1 = {0.f,0.f}, num2 = {0.f,0.f}, den2 = {0.f,0.f};
  const float psx1 = psx, psx4 = 4.f * psx;
  const float psy1 = psy, psy4 = 4.f * psy;
  const bool yA = (unsigned)(gy - 6) < (unsigned)kH;
  const bool yB = (unsigned)(gy - 3) < (unsigned)kH;
  const bool yD = (unsigned)(gy + 3) < (unsigned)kH;
  const bool yE = (unsigned)(gy + 6) < (unsigned)kH;
  const bool xm6 = (unsigned)(gx - 6) < (unsigned)kW;
  const bool xm3 = (unsigned)(gx - 3) < (unsigned)kW;
  const bool xp3 = (unsigned)(gx + 3) < (unsigned)kW;
  const bool xp6 = (unsigned)(gx + 6) < (unsigned)kW;

#define FIN_PAIR(ACC, BL, OFFL, BR, OFFR, SPL, SPR, OKL, OKR)                 \
  { v2f lw = (ACC) + (v2f){ S + (SPL), S + (SPR) };                           \
    v2f w2 = { __expf(lw.x), __expf(lw.y) };                                  \
    if (!interior)                                                            \
      w2 *= (v2f){ (OKL) ? 1.f : 0.f, (OKR) ? 1.f : 0.f };                    \
    den2 += w2;                                                               \
    _Pragma("unroll")                                                         \
    for (int k = 0; k < 3; ++k) {                                             \
      v2f nbk = { (BL)[(OFFL) + k * kLW], (BR)[(OFFR) + k * kLW] };           \
      if (k == 0)      num0 = __builtin_elementwise_fma(w2, nbk, num0);       \
      else if (k == 1) num1 = __builtin_elementwise_fma(w2, nbk, num1);       \
      else             num2 = __builtin_elementwise_fma(w2, nbk, num2);       \
    } }

  FIN_PAIR(a0,  bA, 0, bA, 12, psy4 + psx4, psy4 + psx4, yA && xm6, yA && xp6)
  FIN_PAIR(a1,  bA, 3, bA, 9,  psy4 + psx1, psy4 + psx1, yA && xm3, yA && xp3)
  FIN_PAIR(a2,  bB, 0, bB, 12, psy1 + psx4, psy1 + psx4, yB && xm6, yB && xp6)
  FIN_PAIR(a3,  bB, 3, bB, 9,  psy1 + psx1, psy1 + psx1, yB && xm3, yB && xp3)
  FIN_PAIR(a4,  bC, 0, bC, 12, psx4,        psx4,        xm6,       xp6)
  FIN_PAIR(a5,  bC, 3, bC, 9,  psx1,        psx1,        xm3,       xp3)
  FIN_PAIR(a6,  bD, 0, bD, 12, psy1 + psx4, psy1 + psx4, yD && xm6, yD && xp6)
  FIN_PAIR(a7,  bD, 3, bD, 9,  psy1 + psx1, psy1 + psx1, yD && xm3, yD && xp3)
  FIN_PAIR(a8,  bE, 0, bE, 12, psy4 + psx4, psy4 + psx4, yE && xm6, yE && xp6)
  FIN_PAIR(a9,  bE, 3, bE, 9,  psy4 + psx1, psy4 + psx1, yE && xm3, yE && xp3)
  FIN_PAIR(a10, bA, 6, bB, 6,  psy4,        psy1,        yA,        yB)
  FIN_PAIR(a11, bD, 6, bE, 6,  psy1,        psy4,        yD,        yE)
#undef FIN_PAIR

  // --- center tap is exact: d2 == 0 -> w = 1 ------------------------------
  const float x0 = bC[6], x1 = bC[6 + kLW], x2 = bC[6 + 2 * kLW];
  const float den = den2.x + den2.y + 1.f;
  const float r0 = num0.x + num0.y + x0;
  const float r1 = num1.x + num1.y + x1;
  const float r2 = num2.x + num2.y + x2;

  const float inv = 1.f / den;
  const size_t ob = (size_t)bb * 3 * plane + (size_t)gy * kW + gx;
  out[ob]             = r0 * inv;
  out[ob + plane]     = r1 * inv;
  out[ob + 2 * plane] = r2 * inv;
}

// --- tiny pre-kernel: e[c] = exp(coeffs[c]), c in [0,34) -------------------
__global__ void coeff_exp_kernel(const float* __restrict__ coeffs,
                                 float* __restrict__ e) {
  int i = threadIdx.x;
  if (i < 34) e[i] = __expf(coeffs[i]);
}

extern "C" void kernel_launch(void* const* d_in, const int* in_sizes, int n_in,
                              void* d_out, int out_size, void* d_ws, size_t ws_size,
                              hipStream_t stream) {
  const float* input  = (const float*)d_in[0];   // (2, 66, 720, 1280) f32
  const float* coeffs = (const float*)d_in[1];   // (1, 34, 1, 1)      f32
  float* out = (float*)d_out;                    // (2, 3, 720, 1280)  f32
  float* e   = (float*)d_ws;                     // 34 floats: exp(coeffs)

  coeff_exp_kernel<<<dim3(1), dim3(64), 0, stream>>>(coeffs, e);

  dim3 grid(kW / kTW, kH / kTH, kNB);            // 20 x 45 x 2
  dim3 block(kBX, kBY);                          // 1024 threads = 32 waves
  bilateral_kernel<<<grid, block, 0, stream>>>(input, e, out);
}